// ArcMarginProduct_intertopk_subcenter_18184891531953
// MI455X (gfx1250) — compile-verified
//
#include <hip/hip_runtime.h>

namespace {
constexpr int OUT_F   = 20000;
constexpr int KSUB    = 3;
constexpr int TILE_C  = 2048;
constexpr int TPB     = 256;
constexpr int ROW_IN  = OUT_F * KSUB;            // 60000 floats per row
constexpr unsigned ROW_BYTES = ROW_IN * 4u;      // 240000 bytes per row
constexpr int CHUNK   = TPB * 16;                // 4096 B per async issue (16B/lane x 256)
constexpr int TILE_BYTES = TILE_C * KSUB * 4;    // 24576 B staged per block
constexpr int NCHUNK  = TILE_BYTES / CHUNK;      // 6 async issues

// margin constants (m = 0.2)
constexpr float COS_M = 0.9800665778412416f;     // cos(0.2)
constexpr float SIN_M = 0.19866933079506122f;    // sin(0.2)
constexpr float THR   = -0.9800665778412416f;    // cos(pi - 0.2)
constexpr float MMM   = 0.019933422158758374f;   // 1 + cos(pi - 0.2)
constexpr float SCALE = 32.0f;
}

__global__ __launch_bounds__(TPB) void arc_margin_subcenter_kernel(
    const float* __restrict__ cosine,
    const int*   __restrict__ label,
    float*       __restrict__ out)
{
    __shared__ float smem[TILE_C * KSUB];        // 24 KB tile

    const int tid = threadIdx.x;
    const int b   = blockIdx.y;
    const int c0  = blockIdx.x * TILE_C;

    // 64-bit row base for SADDR; 32-bit per-lane offsets for VADDR (GVS mode).
    const unsigned long long rowBase =
        (unsigned long long)(const void*)(cosine + (size_t)b * ROW_IN);
    // Low 32 bits of the generic address of a __shared__ object == LDS byte offset.
    const unsigned ldsBase = (unsigned)(unsigned long long)(const void*)&smem[0];
    const unsigned tileOff = (unsigned)c0 * (unsigned)(KSUB * 4);

    // Stage the tile via CDNA5 async global->LDS copies: 128b per lane,
    // 4 KB per issue, tracked on ASYNCcnt (no VGPR round-trip).
#pragma unroll
    for (int i = 0; i < NCHUNK; ++i) {
        unsigned goff = tileOff + (unsigned)(i * CHUNK + tid * 16);
        if (goff > ROW_BYTES - 16u) goff = ROW_BYTES - 16u;   // tail clamp (dup loads OK)
        unsigned loff = ldsBase + (unsigned)(i * CHUNK + tid * 16);
        asm volatile("global_load_async_to_lds_b128 %0, %1, %2"
                     :: "v"(loff), "v"(goff), "s"(rowBase)
                     : "memory");
    }
    asm volatile("s_wait_asynccnt 0" ::: "memory");
    __syncthreads();   // other waves' async data must be visible too

    const int lab = label[b];
    float* const orow = out + (size_t)b * OUT_F;

#pragma unroll
    for (int j = 0; j < TILE_C / TPB; ++j) {
        const int cl = tid + j * TPB;      // class within tile
        const int c  = c0 + cl;            // global class id
        if (c < OUT_F) {
            // sub-center max over K=3 (12B-granular reads are cheap from LDS,
            // bank-conflict-free: 3t,3t+1,3t+2 mod 64 is a permutation per wave)
            const float v0 = smem[cl * 3 + 0];
            const float v1 = smem[cl * 3 + 1];
            const float v2 = smem[cl * 3 + 2];
            const float cs = fmaxf(fmaxf(v0, v1), v2);
            float r = cs;                  // phi_mp == cos exactly -> top-k term is identity
            if (c == lab) {
                const float sn  = sqrtf(fmaxf(1.0f - cs * cs, 0.0f));
                const float phi = cs * COS_M - sn * SIN_M;
                r = (cs > THR) ? phi : (cs - MMM);
            }
            __builtin_nontemporal_store(r * SCALE, orow + c);  // streaming store
        }
    }
}

extern "C" void kernel_launch(void* const* d_in, const int* in_sizes, int n_in,
                              void* d_out, int out_size, void* d_ws, size_t ws_size,
                              hipStream_t stream) {
    const float* cosine = (const float*)d_in[0];
    const int*   label  = (const int*)d_in[1];
    float*       out    = (float*)d_out;
    const int B = in_sizes[1];                       // 1024 rows (one label each)
    dim3 grid((OUT_F + TILE_C - 1) / TILE_C, B);     // (10, 1024)
    arc_margin_subcenter_kernel<<<grid, TPB, 0, stream>>>(cosine, label, out);
    (void)n_in; (void)out_size; (void)d_ws; (void)ws_size;
}